// AttnFreeLayer_14663018348826
// MI455X (gfx1250) — compile-verified
//
#include <hip/hip_runtime.h>

// Problem constants (B,S,D fixed by the reference).
#define B_SZ   4
#define S_LEN  8192
#define D_DIM  1024
#define M_TILE 128
#define N_TILE 64
#define NCHUNK (S_LEN / M_TILE)   // 64 chunks along S

typedef float v2f __attribute__((ext_vector_type(2)));
typedef float v8f __attribute__((ext_vector_type(8)));

__device__ __forceinline__ float elu1(float x) {
    // jax.nn.elu(x) + 1  ==  x>0 ? x+1 : exp(x)
    return x > 0.0f ? x + 1.0f : __expf(x);
}

// ---------------------------------------------------------------------------
// Kernel 1: fused triple GEMM (q,k,v projections) via V_WMMA_F32_16X16X4_F32.
// Block = 256 threads = 8 waves covering a 128(M) x 64(N) tile of all three
// GEMMs. Each wave owns a 2x2 arrangement of 16x16 tiles (mg = w>>1 selects
// the 32-row group, ng = w&1 the 32-col group); the second strip/tile is
// addressed with a constant +16*D element offset folded into the load's
// 24-bit immediate. Per 8-K step each lane loads one float4 per matrix
// operand; A/B fragments use the same K permutation (K = 4*half + 2*step +
// vgpr), a bijection applied consistently to both operands -> exact f32 GEMM.
// The K loop is software-pipelined with two fragment sets so 12 loads for
// chunk k+8 are in flight while 24 WMMAs consume chunk k.
// ---------------------------------------------------------------------------
__global__ __launch_bounds__(256) void k1_gemm_fused(
    const float* __restrict__ Xv, const float* __restrict__ Xk,
    const float* __restrict__ Xq,
    const float* __restrict__ Wq, const float* __restrict__ Wk,
    const float* __restrict__ Wv,
    float* __restrict__ q_out,          // d_out staging for q_in
    float* __restrict__ k_in, float* __restrict__ kv_in,
    float* __restrict__ chunkK, float* __restrict__ chunkKV)
{
    const int tid  = threadIdx.x;
    const int lane = tid & 31;
    const int w    = tid >> 5;        // wave id 0..7
    const int l15  = lane & 15;
    const int half = lane >> 4;       // 0 or 1
    const int mg   = w >> 1;          // row group 0..3  (32 rows each)
    const int ng   = w & 1;           // col group 0..1  (32 cols each)

    const int cb    = blockIdx.x * N_TILE;     // column (e) base
    const int chunk = blockIdx.y;              // 128-row chunk along S
    const int b     = blockIdx.z;
    const int r0    = chunk * M_TILE + mg * 32;  // wave's first row strip
    const int c0    = cb + ng * 32;              // wave's first col tile

    // 32-bit per-lane element offsets (max ~33.5M elements) -> saddr+voffset.
    const unsigned aOff = ((unsigned)(b * S_LEN) + (unsigned)(r0 + l15)) * D_DIM
                        + 4u * half;
    const unsigned bOff = (unsigned)(c0 + l15) * D_DIM + 4u * half;

    v8f acc[3][2][2];
#pragma unroll
    for (int g = 0; g < 3; ++g)
#pragma unroll
        for (int mi = 0; mi < 2; ++mi)
#pragma unroll
            for (int ni = 0; ni < 2; ++ni)
                acc[g][mi][ni] = {};

#define LOAD_FRAGS(AF, BF, K)                                                  \
    {                                                                          \
        _Pragma("unroll")                                                      \
        for (int mi = 0; mi < 2; ++mi) {                                       \
            AF[0][mi] = *(const float4*)(Xq + (aOff + (K) + mi * 16 * D_DIM)); \
            AF[1][mi] = *(const float4*)(Xk + (aOff + (K) + mi * 16 * D_DIM)); \
            AF[2][mi] = *(const float4*)(Xv + (aOff + (K) + mi * 16 * D_DIM)); \
        }                                                                      \
        _Pragma("unroll")                                                      \
        for (int ni = 0; ni < 2; ++ni) {                                       \
            BF[0][ni] = *(const float4*)(Wq + (bOff + (K) + ni * 16 * D_DIM)); \
            BF[1][ni] = *(const float4*)(Wk + (bOff + (K) + ni * 16 * D_DIM)); \
            BF[2][ni] = *(const float4*)(Wv + (bOff + (K) + ni * 16 * D_DIM)); \
        }                                                                      \
    }

#define COMPUTE(AF, BF)                                                        \
    {                                                                          \
        _Pragma("unroll")                                                      \
        for (int g = 0; g < 3; ++g)                                            \
            _Pragma("unroll")                                                  \
            for (int mi = 0; mi < 2; ++mi) {                                   \
                v2f a0 = {AF[g][mi].x, AF[g][mi].y};                           \
                v2f a1 = {AF[g][mi].z, AF[g][mi].w};                           \
                _Pragma("unroll")                                              \
                for (int ni = 0; ni < 2; ++ni) {                               \
                    v2f b0 = {BF[g][ni].x, BF[g][ni].y};                       \
                    v2f b1 = {BF[g][ni].z, BF[g][ni].w};                       \
                    acc[g][mi][ni] = __builtin_amdgcn_wmma_f32_16x16x4_f32(    \
                        false, a0, false, b0, (short)0, acc[g][mi][ni],        \
                        false, false);                                         \
                    acc[g][mi][ni] = __builtin_amdgcn_wmma_f32_16x16x4_f32(    \
                        false, a1, false, b1, (short)0, acc[g][mi][ni],        \
                        false, false);                                         \
                }                                                              \
            }                                                                  \
    }

    float4 aC[3][2], bC[3][2], aN[3][2], bN[3][2];
    LOAD_FRAGS(aC, bC, 0u)
    for (int k8 = 0; k8 < D_DIM - 16; k8 += 16) {
        LOAD_FRAGS(aN, bN, (unsigned)(k8 + 8))
        COMPUTE(aC, bC)
        LOAD_FRAGS(aC, bC, (unsigned)(k8 + 16))
        COMPUTE(aN, bN)
    }
    LOAD_FRAGS(aN, bN, (unsigned)(D_DIM - 8))
    COMPUTE(aC, bC)
    COMPUTE(aN, bN)

#undef LOAD_FRAGS
#undef COMPUTE

    // Epilogue. C/D layout: lane l, VGPR j -> row (j + 8*(l>>4)), col (l&15).
    __shared__ float redK[8][2][N_TILE];
    __shared__ float redKV[8][2][N_TILE];

#pragma unroll
    for (int ni = 0; ni < 2; ++ni) {
        const int cl = ng * 32 + ni * 16 + l15;   // local col 0..63
        const int e  = cb + cl;
        float sk = 0.0f, skv = 0.0f;
#pragma unroll
        for (int mi = 0; mi < 2; ++mi) {
#pragma unroll
            for (int j = 0; j < 8; ++j) {
                const int m = r0 + mi * 16 + j + 8 * half;
                const size_t oidx = ((size_t)b * S_LEN + m) * D_DIM + e;
                const float qv  = elu1(acc[0][mi][ni][j]);
                const float kv  = elu1(acc[1][mi][ni][j]);
                const float kvv = kv * acc[2][mi][ni][j];
                q_out[oidx] = qv;
                k_in[oidx]  = kv;
                kv_in[oidx] = kvv;
                sk  += kv;
                skv += kvv;
            }
        }
        redK[w][half][cl]  = sk;
        redKV[w][half][cl] = skv;
    }
    __syncthreads();

    if (tid < N_TILE) {   // deterministic cross-wave reduction (fixed order)
        const int col = tid;
        const int cng = col >> 5;             // which ng group wrote this col
        float sk = 0.0f, skv = 0.0f;
#pragma unroll
        for (int mgi = 0; mgi < 4; ++mgi) {
            const int ww = mgi * 2 + cng;
            sk  += redK[ww][0][col]  + redK[ww][1][col];
            skv += redKV[ww][0][col] + redKV[ww][1][col];
        }
        const size_t cidx = ((size_t)b * NCHUNK + chunk) * D_DIM + (cb + col);
        chunkK[cidx]  = sk;
        chunkKV[cidx] = skv;
    }
}

// ---------------------------------------------------------------------------
// Kernel 2: per-(b,e) serial exclusive scan over the 64 chunk sums (in place).
// 4096 threads, loads coalesced across e.
// ---------------------------------------------------------------------------
__global__ __launch_bounds__(256) void k2_chunk_scan(
    float* __restrict__ chunkK, float* __restrict__ chunkKV)
{
    const int gid = blockIdx.x * blockDim.x + threadIdx.x; // 0 .. B*D-1
    const int b = gid >> 10;
    const int e = gid & (D_DIM - 1);
    size_t idx = (size_t)b * NCHUNK * D_DIM + e;
    float rk = 0.0f, rkv = 0.0f;
    for (int c = 0; c < NCHUNK; ++c) {
        const float vk  = chunkK[idx];
        const float vkv = chunkKV[idx];
        chunkK[idx]  = rk;
        chunkKV[idx] = rkv;
        rk  += vk;
        rkv += vkv;
        idx += D_DIM;
    }
}

// ---------------------------------------------------------------------------
// Kernel 3: within-chunk running scan + final combine. One thread per (b,e)
// per chunk walks 128 rows; all loads/stores coalesced across e. Reads staged
// q_in from d_out and overwrites it in place (same thread, read-then-write).
// Loads are independent (only the running sums are serial) -> unroll for MLP.
// ---------------------------------------------------------------------------
__global__ __launch_bounds__(256) void k3_apply(
    const float* __restrict__ k_in, const float* __restrict__ kv_in,
    const float* __restrict__ chunkK, const float* __restrict__ chunkKV,
    float* __restrict__ out)
{
    const int e     = blockIdx.x * 256 + threadIdx.x;  // 0..D-1
    const int chunk = blockIdx.y;
    const int b     = blockIdx.z;

    const size_t cidx = ((size_t)b * NCHUNK + chunk) * D_DIM + e;
    float rk  = chunkK[cidx];    // exclusive prefixes
    float rkv = chunkKV[cidx];

    size_t idx = ((size_t)b * S_LEN + (size_t)chunk * M_TILE) * D_DIM + e;
#pragma unroll 4
    for (int s = 0; s < M_TILE; ++s) {
        const float kv  = k_in[idx];
        const float kvv = kv_in[idx];
        rk  += kv;    // inclusive k_prefix
        rkv += kvv;   // inclusive kv_prefix
        const float q = out[idx];                 // staged q_in
        out[idx] = q * (kvv + rkv) / rk;          // q_in * (kv_in+kv_prefix)/k_prefix
        idx += D_DIM;
    }
}

// ---------------------------------------------------------------------------
extern "C" void kernel_launch(void* const* d_in, const int* in_sizes, int n_in,
                              void* d_out, int out_size, void* d_ws, size_t ws_size,
                              hipStream_t stream)
{
    (void)in_sizes; (void)n_in; (void)out_size; (void)ws_size;

    // setup_inputs() order: v, k, q, Wq, Wk, Wv (all float32)
    const float* v  = (const float*)d_in[0];
    const float* k  = (const float*)d_in[1];
    const float* q  = (const float*)d_in[2];
    const float* Wq = (const float*)d_in[3];
    const float* Wk = (const float*)d_in[4];
    const float* Wv = (const float*)d_in[5];
    float* out = (float*)d_out;

    const size_t nelem = (size_t)B_SZ * S_LEN * D_DIM;           // 32M floats
    float* k_in    = (float*)d_ws;                               // 128 MB
    float* kv_in   = k_in + nelem;                               // 128 MB
    float* chunkK  = kv_in + nelem;                              // 1 MB
    float* chunkKV = chunkK + (size_t)B_SZ * NCHUNK * D_DIM;     // 1 MB

    dim3 g1(D_DIM / N_TILE, NCHUNK, B_SZ);   // (16, 64, 4)
    k1_gemm_fused<<<g1, 256, 0, stream>>>(v, k, q, Wq, Wk, Wv,
                                          out, k_in, kv_in, chunkK, chunkKV);

    k2_chunk_scan<<<(B_SZ * D_DIM) / 256, 256, 0, stream>>>(chunkK, chunkKV);

    dim3 g3(D_DIM / 256, NCHUNK, B_SZ);      // (4, 64, 4)
    k3_apply<<<g3, 256, 0, stream>>>(k_in, kv_in, chunkK, chunkKV, out);
}